// FeatEmbed_69724499083554
// MI455X (gfx1250) — compile-verified
//
#include <hip/hip_runtime.h>
#include <cstdint>

// FeatEmbed gather: out[b, t*64+j] = tables[t, indices[b,t], j]
// T=16 tables, V=100000 vocab, D=64 dim, B=16384 batch. fp32 in/out, int32 idx.
// Memory-bound: ~129 MiB traffic -> ~5.5us floor @ 23.3 TB/s.

namespace {
constexpr unsigned kT = 16;
constexpr unsigned kV = 100000;
constexpr unsigned kD = 64;
constexpr unsigned kB = 16384;
constexpr unsigned kChunksPerRow = kD / 4;                   // 16 float4 chunks per row
constexpr unsigned kTotalChunks  = kB * kT * kChunksPerRow;  // 4,194,304
constexpr unsigned kBlock = 256;                             // 8 waves (wave32)
constexpr unsigned kIters = 4;                               // chunks per thread
constexpr unsigned kGrid  = kTotalChunks / (kBlock * kIters);// 4096 blocks
constexpr unsigned kGStride = kGrid * kBlock;                // 1,048,576
static_assert(kGrid * kBlock * kIters == kTotalChunks, "exact tiling");
static_assert((kGStride & 255u) == 0u, "t/j4 invariant across iters");
}  // namespace

// Native clang vector types (valid for async builtins and nontemporal ops).
typedef int   v4i __attribute__((vector_size(4 * sizeof(int))));
typedef float v4f __attribute__((vector_size(4 * sizeof(float))));
// The async b128 builtins take pointers to a 16-byte int vector (per hipcc
// diagnostic). AS1 (global) for the memory side, AS3 (local) for the LDS side.
typedef __attribute__((address_space(1))) v4i* g_v4i_ptr;
typedef __attribute__((address_space(3))) v4i* l_v4i_ptr;

#if defined(__AMDGCN__) &&                                            \
    __has_builtin(__builtin_amdgcn_global_load_async_to_lds_b128) && \
    __has_builtin(__builtin_amdgcn_global_store_async_from_lds_b128)
#define FE_USE_ASYNC 1
#else
#define FE_USE_ASYNC 0
#endif

__device__ __forceinline__ void fe_wait_async0() {
#if __has_builtin(__builtin_amdgcn_s_wait_asynccnt)
  __builtin_amdgcn_s_wait_asynccnt(0);
#else
  asm volatile("s_wait_asynccnt 0" ::: "memory");
#endif
}

__global__ __launch_bounds__(kBlock) void feat_embed_gather(
    const float* __restrict__ tables,   // [T, V, D]
    const int* __restrict__ indices,    // [B, T]
    float* __restrict__ out)            // [B, T*D]
{
  const unsigned tid  = threadIdx.x;
  const unsigned gtid = blockIdx.x * kBlock + tid;
  // chunk id c = b*256 + t*16 + j4 ; out chunk address is simply out + 4*c.
  const unsigned j4 = gtid & 15u;         // float4 chunk within D (loop-invariant)
  const unsigned t  = (gtid >> 4) & 15u;  // table id           (loop-invariant)
  const unsigned b0 = gtid >> 8;          // base batch element
  const unsigned tv = t * kV;             // table base row

#if FE_USE_ASYNC
  // memory -> LDS -> memory, data never touches VGPRs (ASYNCcnt-tracked DMA).
  __shared__ char smem[kIters * kBlock * 16];
#pragma unroll
  for (unsigned i = 0; i < kIters; ++i) {
    const unsigned b   = b0 + i * (kGStride >> 8);           // + i*4096
    const unsigned idx = (unsigned)indices[b * kT + t];      // row gather index
    const float* src = tables + (size_t)(tv + idx) * kD + (size_t)j4 * 4u;
    __builtin_amdgcn_global_load_async_to_lds_b128(
        (g_v4i_ptr)src, (l_v4i_ptr)(smem + (i * kBlock + tid) * 16u),
        /*offset=*/0, /*cpol=*/0);
  }
  fe_wait_async0();  // all 4 in-flight row chunks landed in LDS
#pragma unroll
  for (unsigned i = 0; i < kIters; ++i) {
    const unsigned c = gtid + i * kGStride;
    float* dst = out + (size_t)c * 4u;                        // fully coalesced
    __builtin_amdgcn_global_store_async_from_lds_b128(
        (g_v4i_ptr)dst, (l_v4i_ptr)(smem + (i * kBlock + tid) * 16u),
        /*offset=*/0, /*cpol=*/0);
  }
  fe_wait_async0();  // drain stores before wave retires
#else
  // Register fallback: 4 B128 gathered loads in flight, then 4 NT B128 stores.
  v4f v[kIters];
#pragma unroll
  for (unsigned i = 0; i < kIters; ++i) {
    const unsigned b   = b0 + i * (kGStride >> 8);
    const unsigned idx = (unsigned)indices[b * kT + t];
    const v4f* src = (const v4f*)(tables + (size_t)(tv + idx) * kD) + j4;
    v[i] = *src;
  }
#pragma unroll
  for (unsigned i = 0; i < kIters; ++i) {
    const unsigned c = gtid + i * kGStride;
    __builtin_nontemporal_store(v[i], (v4f*)out + c);  // streaming, no reuse
  }
#endif
}

extern "C" void kernel_launch(void* const* d_in, const int* in_sizes, int n_in,
                              void* d_out, int out_size, void* d_ws, size_t ws_size,
                              hipStream_t stream) {
  (void)in_sizes; (void)n_in; (void)out_size; (void)d_ws; (void)ws_size;
  const float* tables  = (const float*)d_in[0];  // [T,V,D] fp32
  const int*   indices = (const int*)d_in[1];    // [B,T] int32
  float*       out     = (float*)d_out;          // [B,T*D] fp32
  feat_embed_gather<<<dim3(kGrid), dim3(kBlock), 0, stream>>>(tables, indices, out);
}